// GCN_10522669875462
// MI455X (gfx1250) — compile-verified
//
#include <hip/hip_runtime.h>
#include <math.h>

typedef __attribute__((ext_vector_type(2))) float v2f;
typedef __attribute__((ext_vector_type(8))) float v8f;

#define LRELU(v) ((v) > 0.0f ? (v) : 0.01f * (v))

// ---------------------------------------------------------------------------
// Zero-fill (workspace/out are poisoned; atomics need clean accumulators)
// ---------------------------------------------------------------------------
__global__ void zero_kernel(float* __restrict__ p, size_t n) {
    size_t i = (size_t)blockIdx.x * blockDim.x + threadIdx.x;
    size_t stride = (size_t)gridDim.x * blockDim.x;
    for (; i < n; i += stride) p[i] = 0.0f;
}

// ---------------------------------------------------------------------------
// Degrees via float atomics (exact: per-node degree << 2^24)
// ---------------------------------------------------------------------------
__global__ void degree_kernel(const int* __restrict__ snd, const int* __restrict__ rcv,
                              float* __restrict__ deg_s, float* __restrict__ deg_r, int E) {
    int e = blockIdx.x * blockDim.x + threadIdx.x;
    if (e < E) {
        atomicAdd(&deg_s[snd[e]], 1.0f);
        atomicAdd(&deg_r[rcv[e]], 1.0f);
    }
}

// deg -> rsqrt(max(deg,1)) in place (deg_s and deg_r are contiguous, n = 2N)
__global__ void invsqrt_kernel(float* __restrict__ deg, int n) {
    int i = blockIdx.x * blockDim.x + threadIdx.x;
    if (i < n) deg[i] = rsqrtf(fmaxf(deg[i], 1.0f));
}

// ---------------------------------------------------------------------------
// Fused GEMM1 -> leaky_relu -> GEMM2 -> (+b2) -> *invs
// One block = 16 node rows. 8 waves, each owns one 16-wide column tile of H=128.
// V_WMMA_F32_16X16X4_F32, K-loop of 32 steps per tile.
// A layout (ISA 7.12.2, 32-bit A 16x4): lane&15 = M, (lane>>4)*2 = K pair.
// B layout (mirror, column-major 4x16):  lane&15 = N, (lane>>4)*2 = K pair.
// C/D layout: vgpr i, lane l -> M = i + 8*(l>>4), N = l&15.
// ---------------------------------------------------------------------------
#define LDST 132  // LDS row stride in floats: (m*132+k)%64 = (4m+k)%64 -> conflict-free

__global__ __launch_bounds__(256) void gemm12_kernel(
    const float* __restrict__ x,  const float* __restrict__ W1, const float* __restrict__ b1,
    const float* __restrict__ W2, const float* __restrict__ b2, const float* __restrict__ invs,
    float* __restrict__ out /* h2 * invs, N x 128 */) {
    const int H = 128;
    __shared__ float axs[16 * LDST];
    __shared__ float h1s[16 * LDST];

    const int rowBase = blockIdx.x * 16;
    const int tid = threadIdx.x;

    // Stage x tile (16 x 128) into LDS, coalesced.
    #pragma unroll
    for (int i = 0; i < 8; ++i) {
        int idx = tid + i * 256;
        int r = idx >> 7, c = idx & 127;
        axs[r * LDST + c] = x[(size_t)(rowBase + r) * H + c];
    }
    __syncthreads();

    const int wave = tid >> 5;
    const int lane = tid & 31;
    const int m  = lane & 15;          // A row within tile
    const int n  = lane & 15;          // B/D column within tile
    const int kk = (lane >> 4) * 2;    // K sub-pair select
    const int colBase = wave * 16;
    const int mBase = (lane >> 4) * 8; // D row base per C/D layout

    // ---- layer 1: h1 = leaky_relu(x @ W1 + b1) ----
    v8f acc = {};
    #pragma unroll 8
    for (int k0 = 0; k0 < H; k0 += 4) {
        v2f a, b;
        a.x = axs[m * LDST + k0 + kk];
        a.y = axs[m * LDST + k0 + kk + 1];
        b.x = W1[(size_t)(k0 + kk) * H + colBase + n];
        b.y = W1[(size_t)(k0 + kk + 1) * H + colBase + n];
        acc = __builtin_amdgcn_wmma_f32_16x16x4_f32(false, a, false, b,
                                                    (short)0, acc, false, false);
    }
    {
        float bias = b1[colBase + n];
        #pragma unroll
        for (int i = 0; i < 8; ++i) {
            float v = acc[i] + bias;
            h1s[(mBase + i) * LDST + colBase + n] = LRELU(v);
        }
    }
    __syncthreads();

    // ---- layer 2: h2 = h1 @ W2 + b2, pre-scaled by invs (conv #1 sender norm) ----
    v8f acc2 = {};
    #pragma unroll 8
    for (int k0 = 0; k0 < H; k0 += 4) {
        v2f a, b;
        a.x = h1s[m * LDST + k0 + kk];
        a.y = h1s[m * LDST + k0 + kk + 1];
        b.x = W2[(size_t)(k0 + kk) * H + colBase + n];
        b.y = W2[(size_t)(k0 + kk + 1) * H + colBase + n];
        acc2 = __builtin_amdgcn_wmma_f32_16x16x4_f32(false, a, false, b,
                                                     (short)0, acc2, false, false);
    }
    {
        float bias = b2[colBase + n];
        #pragma unroll
        for (int i = 0; i < 8; ++i) {
            int row = rowBase + mBase + i;
            out[(size_t)row * H + colBase + n] = (acc2[i] + bias) * invs[row];
        }
    }
}

// ---------------------------------------------------------------------------
// Edge scatter-add, D=128: dst[rcv[e]] += src[snd[e]]
// ---------------------------------------------------------------------------
__global__ void agg128_kernel(const float* __restrict__ src, float* __restrict__ dst,
                              const int* __restrict__ snd, const int* __restrict__ rcv,
                              int E) {
    size_t gid = (size_t)blockIdx.x * blockDim.x + threadIdx.x;
    if (gid >= (size_t)E * 128) return;
    int e = (int)(gid >> 7);
    int c = (int)(gid & 127);
    int s = snd[e], r = rcv[e];
    atomicAdd(&dst[(size_t)r * 128 + c], src[(size_t)s * 128 + c]);
}

// ---------------------------------------------------------------------------
// GEMM3: h3 = (leaky_relu(agg1 * invr) @ Wd + bd) * invs  -> N x 40
// One wave per 16x16 tile; 3 column tiles cover C=40 (masked).
// ---------------------------------------------------------------------------
__global__ __launch_bounds__(256) void gemm3_kernel(
    const float* __restrict__ agg, const float* __restrict__ Wd, const float* __restrict__ bd,
    const float* __restrict__ invr, const float* __restrict__ invs,
    float* __restrict__ out /* N x 40 */, int N) {
    const int H = 128, C = 40, CT = 3;
    const int wave = threadIdx.x >> 5;
    const int lane = threadIdx.x & 31;
    const int tileId = blockIdx.x * 8 + wave;
    const int numTiles = (N / 16) * CT;
    if (tileId >= numTiles) return;           // wave-uniform exit: EXEC stays all-1s

    const int rowBase = (tileId / CT) * 16;
    const int colBase = (tileId % CT) * 16;
    const int m  = lane & 15;
    const int n  = lane & 15;
    const int kk = (lane >> 4) * 2;
    const int col  = colBase + n;
    const int colc = col < C ? col : C - 1;   // clamp address, mask value (no EXEC change)
    const float bmask = col < C ? 1.0f : 0.0f;
    const int rowA = rowBase + m;
    const float ir = invr[rowA];

    v8f acc = {};
    #pragma unroll 8
    for (int k0 = 0; k0 < H; k0 += 4) {
        v2f a, b;
        float a0 = agg[(size_t)rowA * H + k0 + kk] * ir;
        float a1 = agg[(size_t)rowA * H + k0 + kk + 1] * ir;
        a.x = LRELU(a0);
        a.y = LRELU(a1);
        b.x = Wd[(size_t)(k0 + kk) * C + colc] * bmask;
        b.y = Wd[(size_t)(k0 + kk + 1) * C + colc] * bmask;
        acc = __builtin_amdgcn_wmma_f32_16x16x4_f32(false, a, false, b,
                                                    (short)0, acc, false, false);
    }
    if (col < C) {
        float bias = bd[col];
        int mBase = (lane >> 4) * 8;
        #pragma unroll
        for (int i = 0; i < 8; ++i) {
            int r = rowBase + mBase + i;
            out[(size_t)r * C + col] = (acc[i] + bias) * invs[r];
        }
    }
}

// ---------------------------------------------------------------------------
// Edge scatter-add, D=40: 160 threads = 4 edges per block
// ---------------------------------------------------------------------------
__global__ __launch_bounds__(160) void agg40_kernel(
    const float* __restrict__ src, float* __restrict__ dst,
    const int* __restrict__ snd, const int* __restrict__ rcv, int E) {
    int sub = threadIdx.x / 40;
    int c   = threadIdx.x % 40;
    int e = blockIdx.x * 4 + sub;
    if (e >= E) return;
    int s = snd[e], r = rcv[e];
    atomicAdd(&dst[(size_t)r * 40 + c], src[(size_t)s * 40 + c]);
}

// ---------------------------------------------------------------------------
// In-place: row *= invr, then softmax over C=40. One wave per row.
// ---------------------------------------------------------------------------
__global__ __launch_bounds__(256) void softmax_kernel(float* __restrict__ out,
                                                      const float* __restrict__ invr, int N) {
    const int C = 40;
    int wave = threadIdx.x >> 5, lane = threadIdx.x & 31;
    int row = blockIdx.x * 8 + wave;
    if (row >= N) return;
    float ir = invr[row];
    float v0 = out[(size_t)row * C + lane] * ir;                     // lane < 32 < C
    float v1 = (lane + 32 < C) ? out[(size_t)row * C + lane + 32] * ir : -INFINITY;
    float mx = fmaxf(v0, v1);
    #pragma unroll
    for (int o = 16; o > 0; o >>= 1) mx = fmaxf(mx, __shfl_xor(mx, o, 32));
    float e0 = expf(v0 - mx);
    float e1 = (lane + 32 < C) ? expf(v1 - mx) : 0.0f;
    float sum = e0 + e1;
    #pragma unroll
    for (int o = 16; o > 0; o >>= 1) sum += __shfl_xor(sum, o, 32);
    float inv = 1.0f / sum;
    out[(size_t)row * C + lane] = e0 * inv;
    if (lane + 32 < C) out[(size_t)row * C + lane + 32] = e1 * inv;
}

// ---------------------------------------------------------------------------
extern "C" void kernel_launch(void* const* d_in, const int* in_sizes, int n_in,
                              void* d_out, int out_size, void* d_ws, size_t ws_size,
                              hipStream_t stream) {
    const float* x   = (const float*)d_in[0];
    const int*   snd = (const int*)  d_in[1];
    const int*   rcv = (const int*)  d_in[2];
    const float* W1  = (const float*)d_in[3];
    const float* b1  = (const float*)d_in[4];
    const float* W2  = (const float*)d_in[5];
    const float* b2  = (const float*)d_in[6];
    const float* Wd  = (const float*)d_in[7];
    const float* bd  = (const float*)d_in[8];

    const int H = 128, C = 40;
    const int N = in_sizes[0] / H;   // 100000 (multiple of 16)
    const int E = in_sizes[1];       // 600000

    float* ws    = (float*)d_ws;
    float* deg_s = ws;                           // N  -> becomes invs
    float* deg_r = ws + N;                       // N  -> becomes invr
    float* h2s   = deg_r + N;                    // N*128 (sender-scaled h2)
    float* agg1  = h2s + (size_t)N * H;          // N*128
    float* h3s   = h2s;                          // reuse: h2s dead after agg1
    float* outv  = (float*)d_out;                // N*40, doubles as agg2 accumulator

    // 1) zero accumulators (ws/out are poisoned, not re-poisoned between replays)
    zero_kernel<<<1024, 256, 0, stream>>>(deg_s, (size_t)2 * N);
    zero_kernel<<<4096, 256, 0, stream>>>(agg1, (size_t)N * H);
    zero_kernel<<<2048, 256, 0, stream>>>(outv, (size_t)N * C);

    // 2) degrees and symmetric-norm factors
    degree_kernel<<<(E + 255) / 256, 256, 0, stream>>>(snd, rcv, deg_s, deg_r, E);
    invsqrt_kernel<<<(2 * N + 255) / 256, 256, 0, stream>>>(deg_s, 2 * N);

    // 3) fused MLP (GEMM1 -> lrelu -> GEMM2 -> +b2 -> *invs)
    gemm12_kernel<<<N / 16, 256, 0, stream>>>(x, W1, b1, W2, b2, deg_s, h2s);

    // 4) scatter aggregation #1 (D=128)
    {
        size_t total = (size_t)E * 128;
        agg128_kernel<<<(unsigned)((total + 255) / 256), 256, 0, stream>>>(h2s, agg1, snd, rcv, E);
    }

    // 5) decode GEMM: lrelu(agg1*invr) @ Wd + bd, pre-scaled by invs
    {
        int numTiles = (N / 16) * 3;
        gemm3_kernel<<<(numTiles + 7) / 8, 256, 0, stream>>>(agg1, Wd, bd, deg_r, deg_s, h3s, N);
    }

    // 6) scatter aggregation #2 (D=40) into d_out
    agg40_kernel<<<(E + 3) / 4, 160, 0, stream>>>(h3s, outv, snd, rcv, E);

    // 7) receiver norm + softmax, in place
    softmax_kernel<<<(N + 7) / 8, 256, 0, stream>>>(outv, deg_r, N);
}